// PolicyNetwork3x3_8796093022296
// MI455X (gfx1250) — compile-verified
//
#include <hip/hip_runtime.h>
#include <hip/hip_bf16.h>

typedef __attribute__((ext_vector_type(2))) float v2f;
typedef __attribute__((ext_vector_type(8))) float v8f;

// Single-wave fused policy/value network for 3x3 board.
// Matrix layers (fc 32x64, a1 16x32) run on V_WMMA_F32_16X16X4_F32 with the
// weights in the documented fp32 A-matrix layout and the activation K-chunk
// broadcast across all B columns (layout-robust in N).
__global__ __launch_bounds__(32, 1)
void policy3x3_fused(const float* __restrict__ x,
                     const float* __restrict__ conv_w,
                     const float* __restrict__ fc_w,
                     const float* __restrict__ fc_b,
                     const float* __restrict__ a1_w,
                     const float* __restrict__ a1_b,
                     const float* __restrict__ a2_w,
                     const float* __restrict__ a2_b,
                     const float* __restrict__ v1_w,
                     const float* __restrict__ v1_b,
                     const float* __restrict__ v2_w,
                     const float* __restrict__ v2_b,
                     float* __restrict__ out)
{
    __shared__ float h[64];     // conv output (ReLU'd), channel-major flatten
    __shared__ float y1[32];    // fc activations
    __shared__ float a1[16];    // action-head hidden
    __shared__ float logit[9];  // action logits
    __shared__ float vh[8];     // value-head hidden
    __shared__ float tmp32[32]; // raw WMMA accumulator staging

    const int lane = threadIdx.x;   // 0..31, one wave32
    const int l    = lane & 15;     // row within 16-lane half
    const int half = lane >> 4;     // 0: K=0,1  1: K=2,3  (fp32 A layout)

    // Pull the 8KB fc weight block toward the WGP early.
    __builtin_prefetch(fc_w + lane * 16, 0, 3);

    // ---- conv 1x3x3 -> 16x2x2, ReLU; 2 outputs per lane ------------------
    float xr[9];
#pragma unroll
    for (int i = 0; i < 9; ++i) xr[i] = x[i];

#pragma unroll
    for (int r = 0; r < 2; ++r) {
        const int o = lane + 32 * r;          // o = c*4 + (i*2 + j)
        const int c = o >> 2;
        const int p = o & 3;
        const int i0 = p >> 1;
        const int j0 = p & 1;
        const float* w = conv_w + c * 4;
        float acc = xr[i0 * 3 + j0]           * w[0]
                  + xr[i0 * 3 + j0 + 1]       * w[1]
                  + xr[(i0 + 1) * 3 + j0]     * w[2]
                  + xr[(i0 + 1) * 3 + j0 + 1] * w[3];
        h[o] = fmaxf(acc, 0.0f);
    }
    __syncthreads();

    // ---- fc: y1 = relu(fc_w[32x64] @ h + fc_b) via chained WMMA ----------
    // Tile T covers output rows 16T..16T+15; K=64 in 16 chunks of 4.
#pragma unroll
    for (int T = 0; T < 2; ++T) {
        v8f acc = {0.f, 0.f, 0.f, 0.f, 0.f, 0.f, 0.f, 0.f};
#pragma unroll
        for (int k = 0; k < 16; ++k) {
            // A[m][kk] = fc_w[16T+m][4k+kk]; this lane holds row l, K-pair 2*half.
            const float* wr = fc_w + (16 * T + l) * 64 + 4 * k + 2 * half;
            v2f a; a.x = wr[0]; a.y = wr[1];
            // B[kk][n] = h[4k+kk] broadcast over all columns n.
            v2f b; b.x = h[4 * k + 2 * half]; b.y = h[4 * k + 2 * half + 1];
            acc = __builtin_amdgcn_wmma_f32_16x16x4_f32(
                false, a, false, b, (short)0, acc, false, false);
        }
        // Column 0 of D: lane 0 holds M=0..7, lane 16 holds M=8..15.
        if (l == 0) {
#pragma unroll
            for (int m = 0; m < 8; ++m) tmp32[16 * T + 8 * half + m] = acc[m];
        }
        __syncthreads();
    }
    y1[lane] = fmaxf(tmp32[lane] + fc_b[lane], 0.0f);
    __syncthreads();

    // ---- a1: relu(a1_w[16x32] @ y1 + a1_b) via chained WMMA --------------
    {
        v8f acc = {0.f, 0.f, 0.f, 0.f, 0.f, 0.f, 0.f, 0.f};
#pragma unroll
        for (int k = 0; k < 8; ++k) {
            const float* wr = a1_w + l * 32 + 4 * k + 2 * half;
            v2f a; a.x = wr[0]; a.y = wr[1];
            v2f b; b.x = y1[4 * k + 2 * half]; b.y = y1[4 * k + 2 * half + 1];
            acc = __builtin_amdgcn_wmma_f32_16x16x4_f32(
                false, a, false, b, (short)0, acc, false, false);
        }
        if (l == 0) {
#pragma unroll
            for (int m = 0; m < 8; ++m) tmp32[8 * half + m] = acc[m];
        }
    }
    __syncthreads();
    if (lane < 16) a1[lane] = fmaxf(tmp32[lane] + a1_b[lane], 0.0f);
    __syncthreads();

    // ---- tiny tails on VALU ----------------------------------------------
    if (lane < 9) {                 // a2: 9x16 matvec -> logits
        float s = a2_b[lane];
#pragma unroll
        for (int k = 0; k < 16; ++k) s += a1[k] * a2_w[lane * 16 + k];
        logit[lane] = s;
    }
    if (lane < 8) {                 // v1: 8x32 matvec + ReLU
        float s = v1_b[lane];
#pragma unroll
        for (int k = 0; k < 32; ++k) s += y1[k] * v1_w[lane * 32 + k];
        vh[lane] = fmaxf(s, 0.0f);
    }
    __syncthreads();

    if (lane == 0) {
        // value head: tanh(v2_w . vh + v2_b)
        float vv = v2_b[0];
#pragma unroll
        for (int k = 0; k < 8; ++k) vv += vh[k] * v2_w[k];
        out[9] = tanhf(vv);

        // masked softmax over 9 logits (max over ALL logits, mask on exp)
        float maxa = logit[0];
#pragma unroll
        for (int i = 1; i < 9; ++i) maxa = fmaxf(maxa, logit[i]);
        float e[9];
        float s = 0.0f;
#pragma unroll
        for (int i = 0; i < 9; ++i) {
            const float avail = (fabsf(xr[i]) != 1.0f) ? 1.0f : 0.0f;
            e[i] = avail * expf(logit[i] - maxa);
            s += e[i];
        }
        const float inv = 1.0f / s;
#pragma unroll
        for (int i = 0; i < 9; ++i) out[i] = e[i] * inv;
    }
}

extern "C" void kernel_launch(void* const* d_in, const int* in_sizes, int n_in,
                              void* d_out, int out_size, void* d_ws, size_t ws_size,
                              hipStream_t stream) {
    (void)in_sizes; (void)n_in; (void)out_size; (void)d_ws; (void)ws_size;
    const float* x      = (const float*)d_in[0];
    const float* conv_w = (const float*)d_in[1];
    const float* fc_w   = (const float*)d_in[2];
    const float* fc_b   = (const float*)d_in[3];
    const float* a1_w   = (const float*)d_in[4];
    const float* a1_b   = (const float*)d_in[5];
    const float* a2_w   = (const float*)d_in[6];
    const float* a2_b   = (const float*)d_in[7];
    const float* v1_w   = (const float*)d_in[8];
    const float* v1_b   = (const float*)d_in[9];
    const float* v2_w   = (const float*)d_in[10];
    const float* v2_b   = (const float*)d_in[11];
    float* out = (float*)d_out;

    policy3x3_fused<<<1, 32, 0, stream>>>(x, conv_w, fc_w, fc_b,
                                          a1_w, a1_b, a2_w, a2_b,
                                          v1_w, v1_b, v2_w, v2_b, out);
}